// SelectedFrameReconstructor_66185446032284
// MI455X (gfx1250) — compile-verified
//
#include <hip/hip_runtime.h>
#include <stdint.h>

// Problem constants (from reference): S=64, H=294, W=518
#define S_  64
#define H_  294
#define W_  518
#define HW_ (H_ * W_)                 // 152292
#define N_  (S_ * HW_)                // 9,746,688  (== 38073 * 256 exactly)
#define CPF_ ((HW_ + 31) / 32)        // 32-point wave-chunks per frame = 4760

typedef __attribute__((ext_vector_type(2))) float v2f;
typedef __attribute__((ext_vector_type(8))) float v8f;

// ---- order-preserving float <-> uint32 key (for atomicMin z-buffer) ----
__device__ __forceinline__ unsigned fkey(float f) {
  unsigned b = __float_as_uint(f);
  return (b & 0x80000000u) ? ~b : (b | 0x80000000u);
}
__device__ __forceinline__ float fkey_inv(unsigned k) {
  return __uint_as_float((k & 0x80000000u) ? (k ^ 0x80000000u) : ~k);
}

// ----------------------------------------------------------------------
// Wave-cooperative projection of 32 consecutive points of one frame via
// V_WMMA_F32_16X16X4_F32:  D[m][n] = sum_k A[m][k] * B[k][n]
//   A (16x4): point m = [x, y, z, 1]
//   B (4x16): column n (n<4) = row n of M, where M rows 0..2 = K@E, row 3 = E row 2
//   => D[m][0..3] = (u', v', w', z) of point m
// D is routed through LDS so each lane ends up owning one point.
// ----------------------------------------------------------------------
struct Proj {
  float uh, vh, wh, z;  // homogeneous u,v,w and camera depth
  int   s;              // frame
  int   p;              // local pixel/point index in frame (h*W + w)
  bool  inrange;        // lane's point exists (tail-chunk masking)
};

__device__ __forceinline__ void project_wave(const float* __restrict__ wp,
                                             const float* __restrict__ extr,
                                             const float* __restrict__ intr,
                                             float* __restrict__ lds,  // 128 floats per wave
                                             Proj& out) {
  const int tid       = threadIdx.x;
  const int lane      = tid & 31;
  const int waveInBlk = tid >> 5;
  const int waveId    = blockIdx.x * (blockDim.x >> 5) + waveInBlk;
  const int s         = waveId / CPF_;
  const int chunk     = waveId - s * CPF_;
  const int pbase     = chunk * 32;

  const int  myp     = pbase + lane;
  const bool inrange = (myp < HW_);

  // ---------------- B matrix (shared per frame) ----------------
  // b.x: lanes 0-15 -> B[0][n] = Mrow_n[0]; lanes 16-31 -> B[2][n] = Mrow_n[2]
  // b.y: lanes 0-15 -> B[1][n] = Mrow_n[1]; lanes 16-31 -> B[3][n] = Mrow_n[3]
  const float* E = extr + s * 12;  // 3x4 row-major
  const float* K = intr + s * 9;   // 3x3 row-major
  float bx = 0.0f, by = 0.0f;
  {
    const int n  = lane & 15;                // output column
    const int k0 = (lane < 16) ? 0 : 2;      // K-index pair this half-wave holds
    if (n < 3) {
      // Mrow_n = K row n  dot  E columns  (fused projection matrix P = K@E)
      const float ka = K[n * 3 + 0], kb = K[n * 3 + 1], kc = K[n * 3 + 2];
      bx = ka * E[0 * 4 + k0]     + kb * E[1 * 4 + k0]     + kc * E[2 * 4 + k0];
      by = ka * E[0 * 4 + k0 + 1] + kb * E[1 * 4 + k0 + 1] + kc * E[2 * 4 + k0 + 1];
    } else if (n == 3) {
      // Mrow_3 = E row 2  (camera-space z)
      bx = E[2 * 4 + k0];
      by = E[2 * 4 + k0 + 1];
    }
  }
  v2f b;
  b.x = bx;
  b.y = by;

  // ---------------- A matrices: two 16-point groups ----------------
  // lanes 0-15:  a.x = A[m][0]=x, a.y = A[m][1]=y   (m = lane)
  // lanes 16-31: a.x = A[m][2]=z, a.y = A[m][3]=1   (m = lane-16)
  const float* P = wp + (size_t)s * HW_ * 3;
  v2f a0, a1;
  {
    const int m  = lane & 15;
    int q0 = pbase + m;
    int q1 = pbase + 16 + m;
    q0 = (q0 < HW_) ? q0 : (HW_ - 1);  // clamp tail loads (masked later)
    q1 = (q1 < HW_) ? q1 : (HW_ - 1);
    if (lane < 16) {
      a0.x = P[(size_t)q0 * 3 + 0]; a0.y = P[(size_t)q0 * 3 + 1];
      a1.x = P[(size_t)q1 * 3 + 0]; a1.y = P[(size_t)q1 * 3 + 1];
    } else {
      a0.x = P[(size_t)q0 * 3 + 2]; a0.y = 1.0f;
      a1.x = P[(size_t)q1 * 3 + 2]; a1.y = 1.0f;
    }
  }

  // ---------------- WMMA (EXEC is full here; branches above reconverged) ----
  v8f c = {};
  v8f d0 = __builtin_amdgcn_wmma_f32_16x16x4_f32(false, a0, false, b, (short)0, c,
                                                 false, false);
  v8f d1 = __builtin_amdgcn_wmma_f32_16x16x4_f32(false, a1, false, b, (short)0, c,
                                                 false, false);

  // ---------------- Route D through LDS: [32 points][4 comps] per wave -------
  // D layout: VGPR r holds (M=r, N=lane) for lanes 0-15 and (M=r+8, N=lane-16)
  // for lanes 16-31.  Only columns N<4 carry data.
  float* L = lds + waveInBlk * 128;
  const int  col = lane & 15;
  const bool lo  = (lane < 16);
  if (col < 4) {
#pragma unroll
    for (int r = 0; r < 8; ++r) {
      const int m = lo ? r : (r + 8);
      L[m * 4 + col]        = d0[r];
      L[(16 + m) * 4 + col] = d1[r];
    }
  }
  __syncthreads();  // uniform; orders LDS stores before per-lane reads

  out.uh = L[lane * 4 + 0];
  out.vh = L[lane * 4 + 1];
  out.wh = L[lane * 4 + 2];
  out.z  = L[lane * 4 + 3];
  out.s  = s;
  out.p  = myp;
  out.inrange = inrange;
}

// ======================================================================
// Pass 1: initialize outputs (rgb=0, conf=0, depth = key(+inf))
// ======================================================================
__global__ void k_init(float* __restrict__ rgb, unsigned* __restrict__ depthkey,
                       float* __restrict__ conf) {
  const int i = blockIdx.x * blockDim.x + threadIdx.x;
  if (i >= N_) return;
  rgb[i]           = 0.0f;
  rgb[i + N_]      = 0.0f;
  rgb[i + 2 * N_]  = 0.0f;
  conf[i]          = 0.0f;
  depthkey[i]      = 0xFF800000u;  // fkey(+inf)
}

// ======================================================================
// Pass 2: project all points, scatter-min depth (unconditional, like ref)
// ======================================================================
__global__ void k_project_min(const float* __restrict__ wp,
                              const float* __restrict__ extr,
                              const float* __restrict__ intr,
                              unsigned* __restrict__ depthkey) {
  __shared__ float lds[8 * 128];
  Proj pr;
  project_wave(wp, extr, intr, lds, pr);
  if (!pr.inrange) return;

  const float denom = pr.wh + 1e-7f;
  float u = pr.uh / denom;
  float v = pr.vh / denom;
  u = fminf(fmaxf(u, 0.0f), (float)(W_ - 1));
  v = fminf(fmaxf(v, 0.0f), (float)(H_ - 1));
  const int u0 = (int)u;
  const int v0 = (int)v;
  atomicMin(&depthkey[(size_t)pr.s * HW_ + v0 * W_ + u0], fkey(pr.z));
}

// ======================================================================
// Pass 3: recompute projection, hit-test vs depth, bilinear splat rgb+conf
// ======================================================================
__global__ void k_splat(const float* __restrict__ wp,
                        const float* __restrict__ extr,
                        const float* __restrict__ intr,
                        const float* __restrict__ images,
                        const float* __restrict__ confin,
                        const unsigned* __restrict__ depthkey,
                        float* __restrict__ rgb_out,
                        float* __restrict__ conf_out) {
  __shared__ float lds[8 * 128];
  Proj pr;
  project_wave(wp, extr, intr, lds, pr);
  if (!pr.inrange) return;

  const float denom = pr.wh + 1e-7f;
  float u = pr.uh / denom;
  float v = pr.vh / denom;
  const float z = pr.z;
  const bool valid = (z > 0.0f) && (u >= 0.0f) && (u < (float)W_) &&
                     (v >= 0.0f) && (v < (float)H_);
  u = fminf(fmaxf(u, 0.0f), (float)(W_ - 1));
  v = fminf(fmaxf(v, 0.0f), (float)(H_ - 1));
  const int u0 = (int)u;
  const int v0 = (int)v;
  const int u1 = min(u0 + 1, W_ - 1);
  const int v1 = min(v0 + 1, H_ - 1);
  const float du = u - (float)u0;
  const float dv = v - (float)v0;

  const size_t frameBase = (size_t)pr.s * HW_;
  const bool hit = valid && (fkey(z) == depthkey[frameBase + v0 * W_ + u0]);
  if (!hit) return;

  // point's own color / confidence (images is (S,3,H,W) planar)
  const size_t imgBase = (size_t)pr.s * 3 * HW_ + pr.p;
  const float r  = images[imgBase];
  const float g  = images[imgBase + HW_];
  const float bl = images[imgBase + 2 * HW_];
  const float cf = confin[frameBase + pr.p];

  const float w00 = (1.0f - du) * (1.0f - dv);
  const float w01 = (1.0f - du) * dv;
  const float w10 = du * (1.0f - dv);
  const float w11 = du * dv;

  auto splat1 = [&](float wgt, int uu, int vv) {
    if (wgt == 0.0f) return;
    const size_t pix = (size_t)(vv * W_ + uu);
    const size_t q   = (size_t)pr.s * 3 * HW_ + pix;
    atomicAdd(&rgb_out[q],            wgt * r);
    atomicAdd(&rgb_out[q + HW_],      wgt * g);
    atomicAdd(&rgb_out[q + 2 * HW_],  wgt * bl);
    atomicAdd(&conf_out[frameBase + pix], wgt * cf);
  };
  splat1(w00, u0, v0);
  splat1(w01, u0, v1);
  splat1(w10, u1, v0);
  splat1(w11, u1, v1);
}

// ======================================================================
// Pass 4: decode depth keys -> floats, clamp rgb, over-exposure mask
// ======================================================================
__global__ void k_finalize(float* __restrict__ rgb,
                           unsigned* __restrict__ depth,  // keys in, floats out
                           float* __restrict__ mask) {
  const int i = blockIdx.x * blockDim.x + threadIdx.x;
  if (i >= N_) return;
  const float d = fkey_inv(depth[i]);
  const float r  = rgb[i];
  const float g  = rgb[i + N_];
  const float bl = rgb[i + 2 * N_];
  const bool over = (r > 1.0f) || (g > 1.0f) || (bl > 1.0f);
  rgb[i]          = fminf(fmaxf(r, 0.0f), 1.0f);
  rgb[i + N_]     = fminf(fmaxf(g, 0.0f), 1.0f);
  rgb[i + 2 * N_] = fminf(fmaxf(bl, 0.0f), 1.0f);
  ((float*)depth)[i] = d;
  mask[i] = ((d < __builtin_huge_valf()) && !over) ? 1.0f : 0.0f;
}

// ======================================================================
extern "C" void kernel_launch(void* const* d_in, const int* in_sizes, int n_in,
                              void* d_out, int out_size, void* d_ws, size_t ws_size,
                              hipStream_t stream) {
  const float* images = (const float*)d_in[0];  // (S,3,H,W)
  const float* wp     = (const float*)d_in[1];  // (S,H,W,3)
  const float* confin = (const float*)d_in[2];  // (S,H,W)
  const float* extr   = (const float*)d_in[3];  // (S,3,4)
  const float* intr   = (const float*)d_in[4];  // (S,3,3)

  float*    rgb_out  = (float*)d_out;                    // 3*N
  unsigned* depthkey = (unsigned*)((float*)d_out + (size_t)3 * N_);  // N (keys, then floats)
  float*    conf_out = (float*)d_out + (size_t)4 * N_;   // N
  float*    mask_out = (float*)d_out + (size_t)5 * N_;   // N

  const int threads   = 256;
  const int blocksPix = (N_ + threads - 1) / threads;          // 38073
  const int blocksPts = (S_ * CPF_) / (threads / 32);          // 304640 waves / 8 = 38080

  k_init<<<blocksPix, threads, 0, stream>>>(rgb_out, depthkey, conf_out);
  k_project_min<<<blocksPts, threads, 0, stream>>>(wp, extr, intr, depthkey);
  k_splat<<<blocksPts, threads, 0, stream>>>(wp, extr, intr, images, confin,
                                             depthkey, rgb_out, conf_out);
  k_finalize<<<blocksPix, threads, 0, stream>>>(rgb_out, depthkey, mask_out);
}